// EmbeddingIn_16045997818058
// MI455X (gfx1250) — compile-verified
//
#include <hip/hip_runtime.h>
#include <stdint.h>

// ---------------------------------------------------------------------------
// Padded embedding gather for MI455X (gfx1250).
//
// Pure data movement: ~1.08 GB of output writes, W (16.4 MB) is L2-resident.
// HBM-write-bound => use the CDNA5 async DMA path (global<->LDS, ASYNCcnt)
// to stream embedding rows without a VGPR round trip. Loads use TH=RT (keep
// W hot in L2); output stores use TH=NT (streamed, written once).
// ---------------------------------------------------------------------------

#define EMB             256          // floats per embedding row
#define ROW_BYTES       1024         // EMB * 4
#define WAVES_PER_BLOCK 8            // 256 threads, wave32
#define GROUP           4            // output rows handled per wave

// The async-to-LDS builtins take pointers to 16-byte int vectors (per hipcc
// diagnostic: "__attribute__((__vector_size__(4 * sizeof(int)))) int *").
typedef int v4i __attribute__((vector_size(16)));
typedef __attribute__((address_space(1))) v4i gv4i;   // global
typedef __attribute__((address_space(3))) v4i lv4i;   // LDS

#if defined(__gfx1250__) &&                                            \
    __has_builtin(__builtin_amdgcn_global_load_async_to_lds_b128) &&   \
    __has_builtin(__builtin_amdgcn_global_store_async_from_lds_b128)
#define USE_ASYNC_BUILTINS 1
#else
#define USE_ASYNC_BUILTINS 0
#endif

#define CPOL_RT 0   // regular temporal (loads: keep W in L2)
#define CPOL_NT 1   // TH[2:0]=1 non-temporal (stores: streamed output)

// Copy 16 bytes/lane (512 B/wave) global -> LDS, async (ASYNCcnt).
// IMM offset is applied to BOTH the global and LDS addresses by the ISA.
template <int IMM>
__device__ __forceinline__ void async_load_b128(const float* g, uint32_t loff) {
#if USE_ASYNC_BUILTINS
  __builtin_amdgcn_global_load_async_to_lds_b128(
      (gv4i*)(unsigned long long)g, (lv4i*)loff, IMM, CPOL_RT);
#else
  asm volatile("global_load_async_to_lds_b128 %0, %1, off offset:%2"
               :: "v"(loff), "v"(g), "i"(IMM) : "memory");
#endif
}

// Copy 16 bytes/lane LDS -> global, async (ASYNCcnt), non-temporal.
template <int IMM>
__device__ __forceinline__ void async_store_b128(float* g, uint32_t loff) {
#if USE_ASYNC_BUILTINS
  __builtin_amdgcn_global_store_async_from_lds_b128(
      (gv4i*)(unsigned long long)g, (lv4i*)loff, IMM, CPOL_NT);
#else
  asm volatile("global_store_async_from_lds_b128 %0, %1, off offset:%2 th:TH_STORE_NT"
               :: "v"(g), "v"(loff), "i"(IMM) : "memory");
#endif
}

__device__ __forceinline__ void wait_async_0() {
#if defined(__gfx1250__) && __has_builtin(__builtin_amdgcn_s_wait_asynccnt)
  __builtin_amdgcn_s_wait_asynccnt(0);
#else
  asm volatile("s_wait_asynccnt 0" ::: "memory");
#endif
}

// ---------------------------------------------------------------------------
// Async DMA row gather (defined first so it leads the disassembly).
// grid.y = batch b, grid.x covers position groups. One wave per GROUP rows:
//   compute 4 token ids (wave-uniform), async-load 4 rows of W into LDS,
//   s_wait_asynccnt 0, async-store 4 rows to the padded output.
// S_ENDPGM's implicit wait-idle drains the outstanding stores.
// ---------------------------------------------------------------------------
__global__ __launch_bounds__(256) void gather_kernel(
    const float* __restrict__ W, const int* __restrict__ src,
    const int* __restrict__ src_len, const int* __restrict__ offs,
    float* __restrict__ out, int R, int total_tok, int vocab) {
  __shared__ __align__(16) unsigned char smem[WAVES_PER_BLOCK * GROUP * ROW_BYTES];

  const int lane = threadIdx.x & 31;
  const int wave = threadIdx.x >> 5;
  const int b    = blockIdx.y;
  const int p0   = (blockIdx.x * WAVES_PER_BLOCK + wave) * GROUP;
  if (p0 >= R) return;                        // wave-uniform; EXEC stays full

  const int l   = src_len[b];
  const int off = offs[b];
  const int start_id = vocab - 3;             // start / end=+1 / null=+2

  // as3 pointer value == LDS byte offset; flat LDS ptr low 32 bits == offset.
  const uint32_t lds_wave = (uint32_t)(uintptr_t)smem +
                            (uint32_t)(wave * GROUP * ROW_BYTES);
  const uint32_t lane16 = (uint32_t)lane * 16u;

  int nr = 0;
#pragma unroll
  for (int k = 0; k < GROUP; ++k) {
    const int p = p0 + k;
    if (p >= R) break;
    int id;
    if (p == 0) {
      id = start_id;
    } else if (p <= l) {
      int ti = off + p - 1;
      ti = ti < 0 ? 0 : ti;
      ti = ti > total_tok - 1 ? total_tok - 1 : ti;
      id = src[ti];
    } else if (p == l + 1) {
      id = start_id + 1;                      // end token
    } else {
      id = start_id + 2;                      // null token
    }
    const float*  g    = W + (size_t)id * EMB + (size_t)lane * 4;
    const uint32_t loff = lds_wave + (uint32_t)(k * ROW_BYTES) + lane16;
    async_load_b128<0>(g, loff);              // bytes [0, 512) of the row
    async_load_b128<512>(g, loff);            // bytes [512, 1024)
    nr = k + 1;
  }

  wait_async_0();                             // loads done -> LDS valid

  float* orow = out + ((size_t)b * R + p0) * EMB + (size_t)lane * 4;
#pragma unroll
  for (int k = 0; k < GROUP; ++k) {
    if (k >= nr) break;
    const uint32_t loff = lds_wave + (uint32_t)(k * ROW_BYTES) + lane16;
    float* o = orow + (size_t)k * EMB;
    async_store_b128<0>(o, loff);
    async_store_b128<512>(o, loff);
  }
  // implicit S_WAIT_IDLE at S_ENDPGM drains the async stores
}

// ---------------------------------------------------------------------------
// Exclusive prefix sum of src_len -> offs (d_ws), plus the (float)(len + 2)
// tail of the output. One block of 1024 threads; perf-irrelevant.
// ---------------------------------------------------------------------------
__global__ __launch_bounds__(1024) void scan_kernel(const int* __restrict__ len,
                                                    int* __restrict__ offs,
                                                    float* __restrict__ tail,
                                                    int B) {
  __shared__ int part[1024];
  const int t = threadIdx.x;
  const int C = (B + 1023) >> 10;            // elements per thread
  const int i0 = t * C;
  const int i1 = (i0 + C < B) ? (i0 + C) : B;

  int s = 0;
  for (int i = i0; i < i1; ++i) s += len[i];
  part[t] = s;
  __syncthreads();

  // Hillis-Steele inclusive scan over the 1024 partials.
  for (int d = 1; d < 1024; d <<= 1) {
    int x = (t >= d) ? part[t - d] : 0;
    __syncthreads();
    part[t] += x;
    __syncthreads();
  }

  int run = part[t] - s;                     // exclusive prefix at chunk start
  for (int i = i0; i < i1; ++i) {
    offs[i] = run;
    int li = len[i];
    run += li;
    tail[i] = (float)(li + 2);               // second reference output
  }
}

// ---------------------------------------------------------------------------
extern "C" void kernel_launch(void* const* d_in, const int* in_sizes, int n_in,
                              void* d_out, int out_size, void* d_ws, size_t ws_size,
                              hipStream_t stream) {
  const float* W       = (const float*)d_in[0];
  const int*   src     = (const int*)d_in[1];
  const int*   src_len = (const int*)d_in[2];

  const int total = in_sizes[1];
  const int B     = in_sizes[2];
  const int vocab = in_sizes[0] / EMB;                       // 16003
  const int R     = (int)(((long long)out_size - B) /
                          ((long long)B * EMB));             // max_len + 2

  float* out  = (float*)d_out;
  float* tail = out + (size_t)B * (size_t)R * EMB;           // src_len + 2
  int*   offs = (int*)d_ws;                                  // B ints scratch

  scan_kernel<<<1, 1024, 0, stream>>>(src_len, offs, tail, B);

  const int waves_per_seq = (R + GROUP - 1) / GROUP;
  const int bx = (waves_per_seq + WAVES_PER_BLOCK - 1) / WAVES_PER_BLOCK;
  dim3 grid(bx, B, 1);
  gather_kernel<<<grid, 256, 0, stream>>>(W, src, src_len, offs, out,
                                          R, total, vocab);
}